// InstantPolicyModel_86019605004722
// MI455X (gfx1250) — compile-verified
//
#include <hip/hip_runtime.h>

#define NC   65536
#define NA   8192
#define BSZ  128
#define ECC  1048576
#define ECA  262144
#define EAC  262144
#define EAA  131072
#define H    64

typedef __attribute__((ext_vector_type(2))) float v2f;
typedef __attribute__((ext_vector_type(8))) float v8f;

__device__ __forceinline__ v8f wmma4(v2f a, v2f b, v8f c) {
  // D = A(16x4 f32) x B(4x16 f32) + C(16x16 f32)
  return __builtin_amdgcn_wmma_f32_16x16x4_f32(false, a, false, b, (short)0, c,
                                               false, false);
}

__device__ __forceinline__ float gelu_exact(float x) {
  return 0.5f * x * (1.0f + erff(x * 0.7071067811865476f));
}

// -------------------------------------------------------------------------
// h[node,f] = x[node,0]*W[0,f] + x[node,1]*W[1,f] + b[f]
__global__ void embed_kernel(const float* __restrict__ x,
                             const float* __restrict__ W,
                             const float* __restrict__ b,
                             float* __restrict__ h, int n) {
  int tid = blockIdx.x * blockDim.x + threadIdx.x;
  if (tid >= n * H) return;
  int node = tid >> 6, f = tid & 63;
  h[tid] = x[node * 2] * W[f] + x[node * 2 + 1] * W[H + f] + b[f];
}

// -------------------------------------------------------------------------
// Segment-sum of 64-float features over edges + per-dst edge counts.
// 16 threads per edge, 4 floats each.
__global__ void scatter_add_kernel(const float* __restrict__ feat,
                                   const int* __restrict__ src,
                                   const int* __restrict__ dst,
                                   float* __restrict__ sum,
                                   float* __restrict__ cnt, int E) {
  int tid = blockIdx.x * blockDim.x + threadIdx.x;
  int e = tid >> 4;
  if (e >= E) return;
  int c4 = (tid & 15) * 4;
  int s = src[e], d = dst[e];
  const float4 v = *(const float4*)(feat + (size_t)s * H + c4);
  float* o = sum + (size_t)d * H + c4;
  atomicAdd(o + 0, v.x);
  atomicAdd(o + 1, v.y);
  atomicAdd(o + 2, v.z);
  atomicAdd(o + 3, v.w);
  if ((tid & 15) == 0) atomicAdd(cnt + d, 1.0f);
}

// -------------------------------------------------------------------------
// out = relu_opt( (sumA/cntA)@WlA + (sumB/cntB)@WlB + X@(WrA+WrB) + blA+blB )
// One 16-row tile per block (128 threads = 4 waves); wave nt owns columns
// [16*nt, 16*nt+16). fp32 WMMA 16x16x4, K=64 in 16 steps, 3 WMMA per step.
__global__ void __launch_bounds__(128)
sage_out_kernel(const float* __restrict__ sumA, const float* __restrict__ cntA,
                const float* __restrict__ WlA,  const float* __restrict__ blA,
                const float* __restrict__ sumB, const float* __restrict__ cntB,
                const float* __restrict__ WlB,  const float* __restrict__ blB,
                const float* __restrict__ X,    const float* __restrict__ WrA,
                const float* __restrict__ WrB,  float* __restrict__ out,
                int relu) {
  const int base = blockIdx.x * 16;
  const int lane = threadIdx.x & 31;
  const int nt   = threadIdx.x >> 5;   // column tile 0..3
  const int half = lane >> 4;          // 0 -> K pair {0,1}; 1 -> K pair {2,3}
  const int l15  = lane & 15;
  const int rowA = base + l15;         // matrix row carried by this lane (A frags)
  const int col  = nt * 16 + l15;      // output column carried by this lane (B/C)

  const float invA = 1.0f / fmaxf(cntA[rowA], 1.0f);
  const float invB = 1.0f / fmaxf(cntB[rowA], 1.0f);

  const float* sA  = sumA + (size_t)rowA * H + 2 * half;
  const float* sB  = sumB + (size_t)rowA * H + 2 * half;
  const float* sX  = X    + (size_t)rowA * H + 2 * half;
  const float* wlA = WlA + (size_t)(2 * half) * H + col;
  const float* wlB = WlB + (size_t)(2 * half) * H + col;
  const float* wrA = WrA + (size_t)(2 * half) * H + col;
  const float* wrB = WrB + (size_t)(2 * half) * H + col;

  const float bias = blA[col] + blB[col];
  v8f c = {bias, bias, bias, bias, bias, bias, bias, bias};

  for (int k = 0; k < 16; ++k) {
    const int a0 = 4 * k;
    const int w0 = 4 * k * H;
    v2f aM1, aM2, aX, bA, bB, bR;
    aM1[0] = sA[a0] * invA;       aM1[1] = sA[a0 + 1] * invA;
    aM2[0] = sB[a0] * invB;       aM2[1] = sB[a0 + 1] * invB;
    aX[0]  = sX[a0];              aX[1]  = sX[a0 + 1];
    bA[0]  = wlA[w0];             bA[1]  = wlA[w0 + H];
    bB[0]  = wlB[w0];             bB[1]  = wlB[w0 + H];
    bR[0]  = wrA[w0] + wrB[w0];   bR[1]  = wrA[w0 + H] + wrB[w0 + H];
    c = wmma4(aM1, bA, c);
    c = wmma4(aM2, bB, c);
    c = wmma4(aX,  bR, c);
  }

#pragma unroll
  for (int r = 0; r < 8; ++r) {
    float v = c[r];
    if (relu) v = fmaxf(v, 0.0f);
    out[(size_t)(base + r + 8 * half) * H + col] = v;
  }
}

// -------------------------------------------------------------------------
// Sinusoidal time embedding + 2-layer MLP. One block (64 threads) per batch.
__global__ void time_mlp_kernel(const int* __restrict__ timestep,
                                const float* __restrict__ tW1,
                                const float* __restrict__ tb1,
                                const float* __restrict__ tW2,
                                const float* __restrict__ tb2,
                                float* __restrict__ temb_out) {
  __shared__ float te[H];
  __shared__ float hid[2 * H];
  const int b = blockIdx.x;
  const int t = threadIdx.x;
  const float ts = (float)timestep[b];
  const int j = t & 31;
  const float fr = __expf(-9.210340371976184f * (float)j / 31.0f); // log(1e4)
  const float ang = ts * fr;
  te[t] = (t < 32) ? __sinf(ang) : __cosf(ang);
  __syncthreads();
  float acc0 = tb1[t], acc1 = tb1[t + H];
  for (int k = 0; k < H; ++k) {
    const float v = te[k];
    acc0 += v * tW1[k * 2 * H + t];
    acc1 += v * tW1[k * 2 * H + t + H];
  }
  hid[t]     = gelu_exact(acc0);
  hid[t + H] = gelu_exact(acc1);
  __syncthreads();
  float acc = tb2[t];
  for (int k = 0; k < 2 * H; ++k) acc += hid[k] * tW2[k * H + t];
  temb_out[b * H + t] = acc;
}

// -------------------------------------------------------------------------
// fused = [af[i], temb[batch_idx[i]]]; h = gelu(fused@nW1+nb1); out = h@nW2+nb2
// One block (64 threads) per action node.
__global__ void head_kernel(const float* __restrict__ af,
                            const float* __restrict__ temb,
                            const int* __restrict__ batch_idx,
                            const float* __restrict__ nW1,
                            const float* __restrict__ nb1,
                            const float* __restrict__ nW2,
                            const float* __restrict__ nb2,
                            float* __restrict__ out) {
  __shared__ float fused[2 * H];
  __shared__ float hsh[H];
  const int i = blockIdx.x;
  const int t = threadIdx.x;
  const int b = batch_idx[i];
  fused[t]     = af[(size_t)i * H + t];
  fused[t + H] = temb[(size_t)b * H + t];
  __syncthreads();
  float acc = nb1[t];
  for (int k = 0; k < 2 * H; ++k) acc += fused[k] * nW1[k * H + t];
  hsh[t] = gelu_exact(acc);
  __syncthreads();
  if (t < 2) {
    float a = nb2[t];
    for (int k = 0; k < H; ++k) a += hsh[k] * nW2[k * 2 + t];
    out[(size_t)i * 2 + t] = a;
  }
}

// -------------------------------------------------------------------------
extern "C" void kernel_launch(void* const* d_in, const int* in_sizes, int n_in,
                              void* d_out, int out_size, void* d_ws,
                              size_t ws_size, hipStream_t stream) {
  (void)in_sizes; (void)n_in; (void)out_size; (void)ws_size;

  const float* x_context = (const float*)d_in[0];
  const float* x_action  = (const float*)d_in[1];
  const int*   timestep  = (const int*)d_in[2];
  const int*   batch_idx = (const int*)d_in[3];
  const int*   cc_src = (const int*)d_in[4];
  const int*   cc_dst = (const int*)d_in[5];
  const int*   ca_src = (const int*)d_in[6];
  const int*   ca_dst = (const int*)d_in[7];
  const int*   ac_src = (const int*)d_in[8];
  const int*   ac_dst = (const int*)d_in[9];
  const int*   aa_src = (const int*)d_in[10];
  const int*   aa_dst = (const int*)d_in[11];
  const float* W_ctx = (const float*)d_in[12];
  const float* b_ctx = (const float*)d_in[13];
  const float* W_act = (const float*)d_in[14];
  const float* b_act = (const float*)d_in[15];
  const float* tW1 = (const float*)d_in[16];
  const float* tb1 = (const float*)d_in[17];
  const float* tW2 = (const float*)d_in[18];
  const float* tb2 = (const float*)d_in[19];
  const float* c1_Wl = (const float*)d_in[20];
  const float* c1_bl = (const float*)d_in[21];
  const float* c1_Wr = (const float*)d_in[22];
  const float* c2_Wl = (const float*)d_in[23];
  const float* c2_bl = (const float*)d_in[24];
  const float* c2_Wr = (const float*)d_in[25];
  const float* nW1 = (const float*)d_in[26];
  const float* nb1 = (const float*)d_in[27];
  const float* nW2 = (const float*)d_in[28];
  const float* nb2 = (const float*)d_in[29];

  float* wsf = (float*)d_ws;
  // workspace layout (floats)
  float* h_ctx0 = wsf;                     // NC*H
  float* h_act0 = h_ctx0 + (size_t)NC * H; // NA*H
  float* sum_cc = h_act0 + (size_t)NA * H; // NC*H
  float* cnt_cc = sum_cc + (size_t)NC * H; // NC
  float* sum_ac = cnt_cc + NC;             // NC*H
  float* cnt_ac = sum_ac + (size_t)NC * H; // NC
  float* sum_ca = cnt_ac + NC;             // NA*H
  float* cnt_ca = sum_ca + (size_t)NA * H; // NA
  float* sum_aa = cnt_ca + NA;             // NA*H
  float* cnt_aa = sum_aa + (size_t)NA * H; // NA
  float* h_ctx1 = cnt_aa + NA;             // NC*H
  float* h_act1 = h_ctx1 + (size_t)NC * H; // NA*H
  float* actf   = h_act1 + (size_t)NA * H; // NA*H
  float* temb   = actf + (size_t)NA * H;   // BSZ*H

  const size_t aggBytes = (size_t)((char*)h_ctx1 - (char*)sum_cc);
  const size_t agg2Bytes = (size_t)((char*)h_ctx1 - (char*)sum_ca);

  // --- input embeddings ---
  embed_kernel<<<(NC * H) / 256, 256, 0, stream>>>(x_context, W_ctx, b_ctx,
                                                   h_ctx0, NC);
  embed_kernel<<<(NA * H) / 256, 256, 0, stream>>>(x_action, W_act, b_act,
                                                   h_act0, NA);

  // --- layer 1 aggregation ---
  hipMemsetAsync(sum_cc, 0, aggBytes, stream);
  scatter_add_kernel<<<(ECC * 16) / 256, 256, 0, stream>>>(h_ctx0, cc_src,
                                                           cc_dst, sum_cc,
                                                           cnt_cc, ECC);
  scatter_add_kernel<<<(EAC * 16) / 256, 256, 0, stream>>>(h_act0, ac_src,
                                                           ac_dst, sum_ac,
                                                           cnt_ac, EAC);
  scatter_add_kernel<<<(ECA * 16) / 256, 256, 0, stream>>>(h_ctx0, ca_src,
                                                           ca_dst, sum_ca,
                                                           cnt_ca, ECA);
  scatter_add_kernel<<<(EAA * 16) / 256, 256, 0, stream>>>(h_act0, aa_src,
                                                           aa_dst, sum_aa,
                                                           cnt_aa, EAA);

  // --- layer 1 linears (WMMA) + relu ---
  sage_out_kernel<<<NC / 16, 128, 0, stream>>>(
      sum_cc, cnt_cc, c1_Wl + 0 * H * H, c1_bl + 0 * H,
      sum_ac, cnt_ac, c1_Wl + 2 * H * H, c1_bl + 2 * H,
      h_ctx0, c1_Wr + 0 * H * H, c1_Wr + 2 * H * H, h_ctx1, 1);
  sage_out_kernel<<<NA / 16, 128, 0, stream>>>(
      sum_ca, cnt_ca, c1_Wl + 1 * H * H, c1_bl + 1 * H,
      sum_aa, cnt_aa, c1_Wl + 3 * H * H, c1_bl + 3 * H,
      h_act0, c1_Wr + 1 * H * H, c1_Wr + 3 * H * H, h_act1, 1);

  // --- layer 2 aggregation (action dst only) ---
  hipMemsetAsync(sum_ca, 0, agg2Bytes, stream);
  scatter_add_kernel<<<(ECA * 16) / 256, 256, 0, stream>>>(h_ctx1, ca_src,
                                                           ca_dst, sum_ca,
                                                           cnt_ca, ECA);
  scatter_add_kernel<<<(EAA * 16) / 256, 256, 0, stream>>>(h_act1, aa_src,
                                                           aa_dst, sum_aa,
                                                           cnt_aa, EAA);

  // --- layer 2 linears (WMMA), no relu -> action features ---
  sage_out_kernel<<<NA / 16, 128, 0, stream>>>(
      sum_ca, cnt_ca, c2_Wl + 1 * H * H, c2_bl + 1 * H,
      sum_aa, cnt_aa, c2_Wl + 3 * H * H, c2_bl + 3 * H,
      h_act1, c2_Wr + 1 * H * H, c2_Wr + 3 * H * H, actf, 0);

  // --- time embedding MLP ---
  time_mlp_kernel<<<BSZ, H, 0, stream>>>(timestep, tW1, tb1, tW2, tb2, temb);

  // --- fused noise-prediction head ---
  head_kernel<<<NA, H, 0, stream>>>(actf, temb, batch_idx, nW1, nb1, nW2, nb2,
                                    (float*)d_out);
}